// GAE_39470749450257
// MI455X (gfx1250) — compile-verified
//
#include <hip/hip_runtime.h>

// GCN autoencoder: z = BN(Ahat @ (relu(BN(Ahat @ (xW1) )) W2)); out = z Wd + bd
// N=100000 nodes, E=3.2M edges, IN_C=256, HID=64, OUT_C=2. fp32 throughout.
// Memory/scatter bound => fp32 WMMA (V_WMMA_F32_16X16X4_F32) for the one real GEMM.

#define IN_C 256
#define HID  64
#define EPS  1e-5f

typedef __attribute__((ext_vector_type(2))) float v2f;
typedef __attribute__((ext_vector_type(4))) float v4f;
typedef __attribute__((ext_vector_type(8))) float v8f;

__device__ __forceinline__ void atomic_add_f32(float* p, float v) {
  __hip_atomic_fetch_add(p, v, __ATOMIC_RELAXED, __HIP_MEMORY_SCOPE_AGENT);
}

// ---------------------------------------------------------------- init: deg=1 (self loop), zero BN accumulators
__global__ void k_init(float* __restrict__ deg, float* __restrict__ acc1,
                       float* __restrict__ acc2, int N) {
  int i = blockIdx.x * blockDim.x + threadIdx.x;
  if (i < N) deg[i] = 1.0f;
  if (i < 128) acc1[i] = 0.0f;
  if (i < 4) acc2[i] = 0.0f;
}

// ---------------------------------------------------------------- deg[dst] += 1 per edge
__global__ void k_deg(const int* __restrict__ dst, float* __restrict__ deg, int E) {
  int e = blockIdx.x * blockDim.x + threadIdx.x;
  if (e < E) atomic_add_f32(&deg[dst[e]], 1.0f);
}

// ---------------------------------------------------------------- dinv = rsqrt(deg), in place
__global__ void k_rsqrt(float* __restrict__ d, int N) {
  int i = blockIdx.x * blockDim.x + threadIdx.x;
  if (i < N) d[i] = rsqrtf(d[i]);
}

// ---------------------------------------------------------------- xw = x @ W1  via V_WMMA_F32_16X16X4_F32
// One wave computes a 16-row x 64-col strip: 4 accumulator tiles, K=256 in steps of 4.
// A 16x4 f32 frag: lanes 0-15 -> (M=lane, K=k..k+1), lanes 16-31 -> (M=lane-16, K=k+2..k+3).
// B 4x16 frag: v.x = W[krow][col], v.y = W[krow+1][col], krow = k + 2*half, col = 16*tile + l15.
// W1 is staged in LDS k-pair interleaved: sPair[p*64 + col] = {W[2p][col], W[2p+1][col]},
// so every B fragment is ONE contiguous ds_load_b64 into an even VGPR pair (no mov shuffles).
__global__ void __launch_bounds__(256) k_gemm_xw(const float* __restrict__ x,
                                                const float* __restrict__ W1,
                                                float* __restrict__ xw, int N) {
  __shared__ v2f sPair[64 * HID];  // 32 KB: 128 K-rows of W1 as 64 interleaved pairs
  const int lane = threadIdx.x & 31;
  const int wave = threadIdx.x >> 5;
  const int half = lane >> 4;
  const int l15  = lane & 15;

  int arow = blockIdx.x * 128 + wave * 16 + l15;
  int lrow = arow < N ? arow : (N - 1);          // clamp loads; EXEC stays all-1s for WMMA
  const float* xrow = x + (size_t)lrow * IN_C;

  v8f c0 = {}; v8f c1 = {}; v8f c2 = {}; v8f c3 = {};

  for (int kc = 0; kc < IN_C; kc += 128) {
    __syncthreads();
    // stage 128 K-rows of W1, pair-interleaved: sPair[p*64+col] = {W[kc+2p][col], W[kc+2p+1][col]}
    for (int idx = threadIdx.x; idx < 64 * HID; idx += 256) {
      int p = idx >> 6, col = idx & 63;
      v2f w;
      w.x = W1[(size_t)(kc + 2 * p) * HID + col];
      w.y = W1[(size_t)(kc + 2 * p + 1) * HID + col];
      sPair[idx] = w;
    }
    __syncthreads();
#pragma unroll 4
    for (int k = 0; k < 128; k += 4) {
      const int ka = kc + k + half * 2;
      v2f a; a.x = xrow[ka]; a.y = xrow[ka + 1];
      const v2f* bp = sPair + ((k >> 1) + half) * 64 + l15;
      v2f b0 = bp[0];
      v2f b1 = bp[16];
      v2f b2 = bp[32];
      v2f b3 = bp[48];
      c0 = __builtin_amdgcn_wmma_f32_16x16x4_f32(false, a, false, b0, (short)0, c0, false, false);
      c1 = __builtin_amdgcn_wmma_f32_16x16x4_f32(false, a, false, b1, (short)0, c1, false, false);
      c2 = __builtin_amdgcn_wmma_f32_16x16x4_f32(false, a, false, b2, (short)0, c2, false, false);
      c3 = __builtin_amdgcn_wmma_f32_16x16x4_f32(false, a, false, b3, (short)0, c3, false, false);
    }
  }

  // C/D layout: VGPR r -> row M = r + 8*half, col N = lane&15 (+16*tile)
  const int rbase = blockIdx.x * 128 + wave * 16 + half * 8;
#pragma unroll
  for (int r = 0; r < 8; ++r) {
    int row = rbase + r;
    if (row < N) {
      float* o = xw + (size_t)row * HID + l15;
      o[0]  = c0[r];
      o[16] = c1[r];
      o[32] = c2[r];
      o[48] = c3[r];
    }
  }
}

// ---------------------------------------------------------------- out1 = dinv^2 * xw (self loop) + b1
__global__ void k_selfloop1(const float* __restrict__ dinv, const float* __restrict__ xw,
                            const float* __restrict__ b1, float* __restrict__ out1, int N) {
  int i = blockIdx.x * blockDim.x + threadIdx.x;
  if (i < N * HID) {
    int n = i >> 6, c = i & 63;
    float di = dinv[n];
    out1[i] = di * di * xw[i] + b1[c];
  }
}

// ---------------------------------------------------------------- out1[d] += dinv[s]*dinv[d]*xw[s]  (64 lanes/edge)
__global__ void k_scatter1(const int* __restrict__ src, const int* __restrict__ dst,
                           const float* __restrict__ dinv, const float* __restrict__ xw,
                           float* __restrict__ out1, int E) {
  int tid = blockIdx.x * blockDim.x + threadIdx.x;
  int e = tid >> 6, c = tid & 63;
  if (e < E) {
    int s = src[e], d = dst[e];
    float nm = dinv[s] * dinv[d];
    atomic_add_f32(&out1[(size_t)d * HID + c], nm * xw[(size_t)s * HID + c]);
  }
}

// ---------------------------------------------------------------- per-channel sum / sumsq (HID channels)
__global__ void k_stats1(const float* __restrict__ out1, float* __restrict__ acc, int N) {
  int c = threadIdx.x;  // blockDim = 64
  float s = 0.f, q = 0.f;
  for (int r = blockIdx.x; r < N; r += gridDim.x) {
    float v = out1[(size_t)r * HID + c];
    s += v; q += v * v;
  }
  atomic_add_f32(&acc[c], s);
  atomic_add_f32(&acc[HID + c], q);
}

// ---------------------------------------------------------------- acc -> (mu, rsqrt(var+eps))
__global__ void k_finalize(const float* __restrict__ acc, float* __restrict__ mur,
                           int nchan, float invN) {
  int c = threadIdx.x;
  if (c < nchan) {
    float mu = acc[c] * invN;
    float var = acc[nchan + c] * invN - mu * mu;
    mur[c] = mu;
    mur[nchan + c] = rsqrtf(var + EPS);
  }
}

// ---------------------------------------------------------------- h = relu(BN(out1)); hw2 = h @ W2  (fused, 1 thread/node)
__global__ void __launch_bounds__(256) k_bn1_relu_w2(const float* __restrict__ out1,
                                                     const float* __restrict__ mur1,
                                                     const float* __restrict__ g1,
                                                     const float* __restrict__ be1,
                                                     const float* __restrict__ W2,
                                                     float* __restrict__ hw2, int N) {
  __shared__ float sW2[HID * 2], sMu[HID], sRs[HID], sG[HID], sB[HID];
  int t = threadIdx.x;
  if (t < HID * 2) sW2[t] = W2[t];
  if (t < HID) { sMu[t] = mur1[t]; sRs[t] = mur1[HID + t]; sG[t] = g1[t]; sB[t] = be1[t]; }
  __syncthreads();
  int n = blockIdx.x * blockDim.x + t;
  if (n >= N) return;
  const v4f* row = (const v4f*)(out1 + (size_t)n * HID);
  float a0 = 0.f, a1 = 0.f;
#pragma unroll
  for (int i = 0; i < HID / 4; ++i) {
    v4f v = row[i];
#pragma unroll
    for (int u = 0; u < 4; ++u) {
      int c = i * 4 + u;
      float h = (v[u] - sMu[c]) * sRs[c] * sG[c] + sB[c];
      h = h > 0.f ? h : 0.f;
      a0 += h * sW2[2 * c];
      a1 += h * sW2[2 * c + 1];
    }
  }
  hw2[(size_t)n * 2]     = a0;
  hw2[(size_t)n * 2 + 1] = a1;
}

// ---------------------------------------------------------------- z1 = dinv^2 * hw2 (self loop) + b2
__global__ void k_selfloop2(const float* __restrict__ dinv, const float* __restrict__ hw2,
                            const float* __restrict__ b2, float* __restrict__ z1, int N) {
  int n = blockIdx.x * blockDim.x + threadIdx.x;
  if (n < N) {
    float di = dinv[n];
    float nm = di * di;
    z1[(size_t)n * 2]     = nm * hw2[(size_t)n * 2]     + b2[0];
    z1[(size_t)n * 2 + 1] = nm * hw2[(size_t)n * 2 + 1] + b2[1];
  }
}

// ---------------------------------------------------------------- z1[d] += norm * hw2[s]  (2 channels/edge)
__global__ void k_scatter2(const int* __restrict__ src, const int* __restrict__ dst,
                           const float* __restrict__ dinv, const float* __restrict__ hw2,
                           float* __restrict__ z1, int E) {
  int e = blockIdx.x * blockDim.x + threadIdx.x;
  if (e < E) {
    int s = src[e], d = dst[e];
    float nm = dinv[s] * dinv[d];
    atomic_add_f32(&z1[(size_t)d * 2],     nm * hw2[(size_t)s * 2]);
    atomic_add_f32(&z1[(size_t)d * 2 + 1], nm * hw2[(size_t)s * 2 + 1]);
  }
}

// ---------------------------------------------------------------- BN2 stats: LDS tree reduce, 4 atomics/block
__global__ void __launch_bounds__(256) k_stats2(const float* __restrict__ z1,
                                                float* __restrict__ acc2, int N) {
  __shared__ float red[256][4];
  float s0 = 0.f, q0 = 0.f, s1 = 0.f, q1 = 0.f;
  for (int n = blockIdx.x * blockDim.x + threadIdx.x; n < N; n += gridDim.x * blockDim.x) {
    float a = z1[(size_t)n * 2];
    float b = z1[(size_t)n * 2 + 1];
    s0 += a; q0 += a * a; s1 += b; q1 += b * b;
  }
  red[threadIdx.x][0] = s0; red[threadIdx.x][1] = q0;
  red[threadIdx.x][2] = s1; red[threadIdx.x][3] = q1;
  __syncthreads();
  for (int off = 128; off > 0; off >>= 1) {
    if (threadIdx.x < off)
      for (int w = 0; w < 4; ++w) red[threadIdx.x][w] += red[threadIdx.x + off][w];
    __syncthreads();
  }
  if (threadIdx.x == 0) {
    atomic_add_f32(&acc2[0], red[0][0]);  // sum ch0
    atomic_add_f32(&acc2[1], red[0][2]);  // sum ch1
    atomic_add_f32(&acc2[2], red[0][1]);  // sumsq ch0
    atomic_add_f32(&acc2[3], red[0][3]);  // sumsq ch1
  }
}

// ---------------------------------------------------------------- out[n,:] = BN2(z1[n]) @ Wd + bd   (1 block/node)
__global__ void __launch_bounds__(256) k_decode(const float* __restrict__ z1,
                                                const float* __restrict__ mur2,
                                                const float* __restrict__ g2,
                                                const float* __restrict__ be2,
                                                const float* __restrict__ Wd,
                                                const float* __restrict__ bd,
                                                float* __restrict__ out, int N) {
  __shared__ float zs[2];
  const int n = blockIdx.x;
  if (threadIdx.x < 2) {
    int c = threadIdx.x;
    float v = z1[(size_t)n * 2 + c];
    zs[c] = (v - mur2[c]) * mur2[2 + c] * g2[c] + be2[c];
  }
  __syncthreads();
  const int j = threadIdx.x;
  out[(size_t)n * IN_C + j] = zs[0] * Wd[j] + zs[1] * Wd[IN_C + j] + bd[j];
}

extern "C" void kernel_launch(void* const* d_in, const int* in_sizes, int n_in,
                              void* d_out, int out_size, void* d_ws, size_t ws_size,
                              hipStream_t stream) {
  const float* x   = (const float*)d_in[0];
  const int*   ei  = (const int*)d_in[1];   // edge_index [2,E], row0=src, row1=dst (per bridge doc: integer -> int32)
  const float* W1  = (const float*)d_in[2];
  const float* b1  = (const float*)d_in[3];
  const float* g1  = (const float*)d_in[4];
  const float* be1 = (const float*)d_in[5];
  const float* W2  = (const float*)d_in[6];
  const float* b2  = (const float*)d_in[7];
  const float* g2  = (const float*)d_in[8];
  const float* be2 = (const float*)d_in[9];
  const float* Wd  = (const float*)d_in[10];
  const float* bd  = (const float*)d_in[11];
  float* out = (float*)d_out;

  const int N = in_sizes[0] / IN_C;
  const int E = in_sizes[1] / 2;
  const int* src = ei;
  const int* dst = ei + E;

  // workspace carve-up (floats): dinv | xw | out1 | hw2 | z1 | acc1 | mur1 | acc2 | mur2
  float* ws = (float*)d_ws;
  size_t o = 0;
  float* dinv = ws + o; o += (size_t)N;
  float* xw   = ws + o; o += (size_t)N * HID;
  float* out1 = ws + o; o += (size_t)N * HID;
  float* hw2  = ws + o; o += (size_t)N * 2;
  float* z1   = ws + o; o += (size_t)N * 2;
  float* acc1 = ws + o; o += 128;
  float* mur1 = ws + o; o += 128;
  float* acc2 = ws + o; o += 4;
  float* mur2 = ws + o; o += 4;

  const int TB = 256;
  k_init<<<(N + TB - 1) / TB, TB, 0, stream>>>(dinv, acc1, acc2, N);
  k_deg<<<(E + TB - 1) / TB, TB, 0, stream>>>(dst, dinv, E);
  k_rsqrt<<<(N + TB - 1) / TB, TB, 0, stream>>>(dinv, N);
  k_gemm_xw<<<(N + 127) / 128, TB, 0, stream>>>(x, W1, xw, N);
  k_selfloop1<<<((size_t)N * HID + TB - 1) / TB, TB, 0, stream>>>(dinv, xw, b1, out1, N);
  k_scatter1<<<(unsigned)(((long long)E * HID + TB - 1) / TB), TB, 0, stream>>>(src, dst, dinv, xw, out1, E);
  k_stats1<<<1024, 64, 0, stream>>>(out1, acc1, N);
  k_finalize<<<1, 64, 0, stream>>>(acc1, mur1, HID, 1.0f / (float)N);
  k_bn1_relu_w2<<<(N + TB - 1) / TB, TB, 0, stream>>>(out1, mur1, g1, be1, W2, hw2, N);
  k_selfloop2<<<(N + TB - 1) / TB, TB, 0, stream>>>(dinv, hw2, b2, z1, N);
  k_scatter2<<<(E + TB - 1) / TB, TB, 0, stream>>>(src, dst, dinv, hw2, z1, E);
  k_stats2<<<512, TB, 0, stream>>>(z1, acc2, N);
  k_finalize<<<1, 64, 0, stream>>>(acc2, mur2, 2, 1.0f / (float)N);
  k_decode<<<N, TB, 0, stream>>>(z1, mur2, g2, be2, Wd, bd, out, N);
}